// GINe_85263690760421
// MI455X (gfx1250) — compile-verified
//
#include <hip/hip_runtime.h>

#define N_NODES 100000
#define N_EDGES 640000
#define NLAYER  2

typedef float v2f __attribute__((ext_vector_type(2)));
typedef float v8f __attribute__((ext_vector_type(8)));

__device__ __forceinline__ v8f wmma_f32(v2f a, v2f b, v8f c) {
  // D = A(16x4, f32) * B(4x16, f32) + C(16x16, f32)
  return __builtin_amdgcn_wmma_f32_16x16x4_f32(false, a, false, b, (short)0, c, false, false);
}

__device__ __forceinline__ void atomic_add_f32(float* p, float v) {
  unsafeAtomicAdd(p, v);  // -> global_atomic_add_f32
}

// async global -> LDS copy, 4 bytes per lane, tracked by ASYNCcnt
__device__ __forceinline__ void async_g2l_b32(float* lds_ptr, const float* gptr) {
  unsigned int l = (unsigned int)(uintptr_t)lds_ptr;        // low 32b = LDS offset
  unsigned long long g = (unsigned long long)(uintptr_t)gptr;
  asm volatile("global_load_async_to_lds_b32 %0, %1, off" :: "v"(l), "v"(g) : "memory");
}
__device__ __forceinline__ void wait_async0() {
  asm volatile("s_wait_asynccnt 0x0" ::: "memory");
}

__device__ __forceinline__ v2f lds_ld2(const float* p) { return *(const v2f*)p; }

#define WSTRIDE 68   // padded LDS stride: bank = (4*col + k) & 63, conflict-free, 8B aligned

// ---------------------------------------------------------------------------
// zero fill (vectorized, n must be multiple of 4)
// ---------------------------------------------------------------------------
__global__ void fill_zero_kernel(float4* __restrict__ p, size_t n4) {
  size_t i = (size_t)blockIdx.x * blockDim.x + threadIdx.x;
  size_t step = (size_t)gridDim.x * blockDim.x;
  float4 z = make_float4(0.f, 0.f, 0.f, 0.f);
  for (; i < n4; i += step) p[i] = z;
}

// ---------------------------------------------------------------------------
// out[M,64] = in[M,K] @ W[K,64] + bias  (embeddings)
// ---------------------------------------------------------------------------
template<int K>
__global__ void gemm64_kernel(const float* __restrict__ in, const float* __restrict__ W,
                              const float* __restrict__ bias, float* __restrict__ out, int M) {
  __shared__ float Wt[64 * WSTRIDE];
  int tid = threadIdx.x;
  {  // async stage: Wt[c*WSTRIDE + r] = W[r*64 + c]
    int c = tid & 63, r0 = tid >> 6;
#pragma unroll
    for (int rr = 0; rr < K / 4; ++rr) {
      int r = rr * 4 + r0;
      async_g2l_b32(&Wt[c * WSTRIDE + r], W + (size_t)r * 64 + c);
    }
  }
  int wave = blockIdx.x * (blockDim.x >> 5) + (tid >> 5);
  int lane = tid & 31;
  int half = lane >> 4, ml = lane & 15;
  int m0 = wave * 16;
  bool active = m0 < M;  // wave-uniform -> EXEC all-ones around WMMA

  constexpr int KS = K / 4;
  v2f a[KS];
  if (active) {  // overlap A loads with async weight DMA
    unsigned ao = (unsigned)(m0 + ml) * K;
#pragma unroll
    for (int ks = 0; ks < KS; ++ks) {
      int k = ks * 4 + 2 * half;
      a[ks].x = in[ao + k]; a[ks].y = in[ao + k + 1];
    }
  }
  wait_async0();
  __syncthreads();
  if (active) {
#pragma unroll
    for (int nt = 0; nt < 4; ++nt) {
      int col = nt * 16 + ml;
      v8f acc = {};
#pragma unroll
      for (int ks = 0; ks < KS; ++ks) {
        v2f b = lds_ld2(&Wt[col * WSTRIDE + ks * 4 + 2 * half]);
        acc = wmma_f32(a[ks], b, acc);
      }
      float bv = bias[col];
#pragma unroll
      for (int r = 0; r < 8; ++r)
        out[(size_t)(m0 + r + 8 * half) * 64 + col] = acc[r] + bv;
    }
  }
}

// ---------------------------------------------------------------------------
// msg = relu(x[src] + e@W + b); agg[dst] += msg
// x gathers issued BEFORE the WMMA chain so random-access latency overlaps it
// ---------------------------------------------------------------------------
__global__ void msg_scatter_kernel(const float* __restrict__ x, const float* __restrict__ e,
                                   const int* __restrict__ src, const int* __restrict__ dst,
                                   const float* __restrict__ W, const float* __restrict__ bias,
                                   float* __restrict__ agg) {
  __shared__ float Wt[64 * WSTRIDE];
  int tid = threadIdx.x;
  {
    int c = tid & 63, r0 = tid >> 6;
#pragma unroll
    for (int rr = 0; rr < 16; ++rr) {
      int r = rr * 4 + r0;
      async_g2l_b32(&Wt[c * WSTRIDE + r], W + (size_t)r * 64 + c);
    }
  }
  int wave = blockIdx.x * (blockDim.x >> 5) + (tid >> 5);
  int lane = tid & 31;
  int half = lane >> 4, ml = lane & 15;
  int m0 = wave * 16;
  bool active = m0 < N_EDGES;

  v2f a[16];
  unsigned doff[8];
  float xv[4][8];
  if (active) {
    unsigned eo = (unsigned)(m0 + ml) * 64;
#pragma unroll
    for (int ks = 0; ks < 16; ++ks) {
      int k = ks * 4 + 2 * half;
      a[ks].x = e[eo + k]; a[ks].y = e[eo + k + 1];
    }
    unsigned soff[8];
#pragma unroll
    for (int r = 0; r < 8; ++r) {
      int row = m0 + r + 8 * half;
      soff[r] = (unsigned)src[row] * 64u;
      doff[r] = (unsigned)dst[row] * 64u;
    }
#pragma unroll
    for (int nt = 0; nt < 4; ++nt)
#pragma unroll
      for (int r = 0; r < 8; ++r)
        xv[nt][r] = x[soff[r] + (unsigned)(nt * 16 + ml)];
  }
  wait_async0();
  __syncthreads();
  if (active) {
#pragma unroll
    for (int nt = 0; nt < 4; ++nt) {
      int col = nt * 16 + ml;
      v8f acc = {};
#pragma unroll
      for (int ks = 0; ks < 16; ++ks) {
        v2f b = lds_ld2(&Wt[col * WSTRIDE + ks * 4 + 2 * half]);
        acc = wmma_f32(a[ks], b, acc);
      }
      float bv = bias[col];
#pragma unroll
      for (int r = 0; r < 8; ++r) {
        float v = xv[nt][r] + acc[r] + bv;
        v = fmaxf(v, 0.f);
        atomic_add_f32(&agg[doff[r] + (unsigned)col], v);
      }
    }
  }
}

// ---------------------------------------------------------------------------
// conv fused: h = relu((x+agg)@W1 + b1) @ W2 + b2
// + per-channel sum/sumsq reduction for BatchNorm (ds_add_f32 + global atomics)
// ---------------------------------------------------------------------------
__global__ void conv_fused_kernel(const float* __restrict__ x, const float* __restrict__ agg,
                                  const float* __restrict__ W1, const float* __restrict__ b1,
                                  const float* __restrict__ W2, const float* __restrict__ b2,
                                  float* __restrict__ h, float* __restrict__ stats) {
  __shared__ float tlds[8][16 * WSTRIDE];
  __shared__ float W2t[64 * WSTRIDE];
  __shared__ float ssum[64];
  __shared__ float ssq[64];
  int tid = threadIdx.x;
  {
    int c = tid & 63, r0 = tid >> 6;
#pragma unroll
    for (int rr = 0; rr < 16; ++rr) {
      int r = rr * 4 + r0;
      async_g2l_b32(&W2t[c * WSTRIDE + r], W2 + (size_t)r * 64 + c);
    }
  }
  if (tid < 64) { ssum[tid] = 0.f; ssq[tid] = 0.f; }

  int wib = tid >> 5;
  int wave = blockIdx.x * (blockDim.x >> 5) + wib;
  int lane = tid & 31;
  int half = lane >> 4, ml = lane & 15;
  int m0 = wave * 16;
  bool active = m0 < N_NODES;
  float* tl = tlds[wib];

  if (active) {
    unsigned ao = (unsigned)(m0 + ml) * 64u;
    v2f a[16];
#pragma unroll
    for (int ks = 0; ks < 16; ++ks) {
      int k = ks * 4 + 2 * half;
      a[ks].x = x[ao + k] + agg[ao + k];
      a[ks].y = x[ao + k + 1] + agg[ao + k + 1];
    }
#pragma unroll
    for (int nt = 0; nt < 4; ++nt) {
      int col = nt * 16 + ml;
      v8f acc = {};
#pragma unroll
      for (int ks = 0; ks < 16; ++ks) {
        int k = ks * 4 + 2 * half;
        v2f b;
        b.x = W1[(unsigned)k * 64 + col];
        b.y = W1[(unsigned)(k + 1) * 64 + col];
        acc = wmma_f32(a[ks], b, acc);
      }
      float bv = b1[col];
#pragma unroll
      for (int r = 0; r < 8; ++r)
        tl[(r + 8 * half) * WSTRIDE + col] = fmaxf(acc[r] + bv, 0.f);
    }
  }
  wait_async0();
  __syncthreads();
  if (active) {
    v2f a[16];
#pragma unroll
    for (int ks = 0; ks < 16; ++ks)
      a[ks] = lds_ld2(&tl[ml * WSTRIDE + ks * 4 + 2 * half]);
#pragma unroll
    for (int nt = 0; nt < 4; ++nt) {
      int col = nt * 16 + ml;
      v8f acc = {};
#pragma unroll
      for (int ks = 0; ks < 16; ++ks) {
        v2f b = lds_ld2(&W2t[col * WSTRIDE + ks * 4 + 2 * half]);
        acc = wmma_f32(a[ks], b, acc);
      }
      float bv = b2[col];
      float s = 0.f, q = 0.f;
#pragma unroll
      for (int r = 0; r < 8; ++r) {
        float v = acc[r] + bv;
        h[(size_t)(m0 + r + 8 * half) * 64 + col] = v;
        s += v; q += v * v;
      }
      atomicAdd(&ssum[col], s);  // ds_add_f32
      atomicAdd(&ssq[col], q);
    }
  }
  __syncthreads();
  if (tid < 64) {
    atomicAdd(&stats[tid], ssum[tid]);
    atomicAdd(&stats[64 + tid], ssq[tid]);
  }
}

// ---------------------------------------------------------------------------
// xout = (x + relu(gamma*(h-mu)*rsqrt(var+eps)+beta)) / 2   (b128 vectorized)
// ---------------------------------------------------------------------------
__global__ void bn_apply_kernel(const float* __restrict__ x, const float* __restrict__ h,
                                const float* __restrict__ stats,
                                const float* __restrict__ gamma, const float* __restrict__ beta,
                                float* __restrict__ xout) {
  size_t i = (size_t)blockIdx.x * blockDim.x + threadIdx.x;
  if (i >= (size_t)N_NODES * 16) return;
  size_t base = i * 4;
  int c = (int)(base & 63);
  const float invN = 1.f / (float)N_NODES;
  float4 hv = *(const float4*)(h + base);
  float4 xv = *(const float4*)(x + base);
  float4 o;
  const float* hp = (const float*)&hv;
  const float* xp = (const float*)&xv;
  float* op = (float*)&o;
#pragma unroll
  for (int j = 0; j < 4; ++j) {
    float mu  = stats[c + j] * invN;
    float var = stats[64 + c + j] * invN - mu * mu;
    float inv = rsqrtf(var + 1e-5f);
    float hn = gamma[c + j] * (hp[j] - mu) * inv + beta[c + j];
    op[j] = (xp[j] + fmaxf(hn, 0.f)) * 0.5f;
  }
  *(float4*)(xout + base) = o;
}

// ---------------------------------------------------------------------------
// e += 0.5 * ( relu([x_src|x_dst|e] @ W1[192,64] + b1) @ W2[64,64] + b2 )
// ---------------------------------------------------------------------------
__global__ void edge_update_kernel(const float* __restrict__ x, float* e,
                                   const int* __restrict__ src, const int* __restrict__ dst,
                                   const float* __restrict__ W1, const float* __restrict__ b1,
                                   const float* __restrict__ W2, const float* __restrict__ b2) {
  __shared__ float tlds[8][16 * WSTRIDE];
  __shared__ float W2t[64 * WSTRIDE];
  int tid = threadIdx.x;
  {
    int c = tid & 63, r0 = tid >> 6;
#pragma unroll
    for (int rr = 0; rr < 16; ++rr) {
      int r = rr * 4 + r0;
      async_g2l_b32(&W2t[c * WSTRIDE + r], W2 + (size_t)r * 64 + c);
    }
  }
  int wib = tid >> 5;
  int wave = blockIdx.x * (blockDim.x >> 5) + wib;
  int lane = tid & 31;
  int half = lane >> 4, ml = lane & 15;
  int m0 = wave * 16;
  bool active = m0 < N_EDGES;
  float* tl = tlds[wib];

  if (active) {
    int rowa = m0 + ml;
    unsigned so = (unsigned)src[rowa] * 64u;
    unsigned dof = (unsigned)dst[rowa] * 64u;
    unsigned eo = (unsigned)rowa * 64u;
    v8f acc[4] = {};
#pragma unroll
    for (int kc = 0; kc < 3; ++kc) {
      v2f a[16];
#pragma unroll
      for (int ks = 0; ks < 16; ++ks) {
        int k = ks * 4 + 2 * half;
        unsigned off = (kc == 0) ? (so + k) : ((kc == 1) ? (dof + k) : (eo + k));
        const float* p = (kc == 2) ? e : x;
        a[ks].x = p[off]; a[ks].y = p[off + 1];
      }
#pragma unroll
      for (int nt = 0; nt < 4; ++nt) {
        int col = nt * 16 + ml;
#pragma unroll
        for (int ks = 0; ks < 16; ++ks) {
          int k = kc * 64 + ks * 4 + 2 * half;
          v2f b;
          b.x = W1[(unsigned)k * 64 + col];
          b.y = W1[(unsigned)(k + 1) * 64 + col];
          acc[nt] = wmma_f32(a[ks], b, acc[nt]);
        }
      }
    }
#pragma unroll
    for (int nt = 0; nt < 4; ++nt) {
      int col = nt * 16 + ml;
      float bv = b1[col];
#pragma unroll
      for (int r = 0; r < 8; ++r)
        tl[(r + 8 * half) * WSTRIDE + col] = fmaxf(acc[nt][r] + bv, 0.f);
    }
  }
  wait_async0();
  __syncthreads();
  if (active) {
    v2f a[16];
#pragma unroll
    for (int ks = 0; ks < 16; ++ks)
      a[ks] = lds_ld2(&tl[ml * WSTRIDE + ks * 4 + 2 * half]);
#pragma unroll
    for (int nt = 0; nt < 4; ++nt) {
      int col = nt * 16 + ml;
      v8f acc = {};
#pragma unroll
      for (int ks = 0; ks < 16; ++ks) {
        v2f b = lds_ld2(&W2t[col * WSTRIDE + ks * 4 + 2 * half]);
        acc = wmma_f32(a[ks], b, acc);
      }
      float bv = b2[col];
#pragma unroll
      for (int r = 0; r < 8; ++r) {
        size_t idx = (size_t)(m0 + r + 8 * half) * 64 + col;
        e[idx] = e[idx] + (acc[r] + bv) * 0.5f;
      }
    }
  }
}

// ---------------------------------------------------------------------------
// readout: er = [relu(x_src)|relu(x_dst)|e] (192) -> 50 -> 25 -> 1
// ---------------------------------------------------------------------------
#define H2S 36
__global__ void readout_kernel(const float* __restrict__ x, const float* __restrict__ e,
                               const int* __restrict__ src, const int* __restrict__ dst,
                               const float* __restrict__ w1, const float* __restrict__ b1,
                               const float* __restrict__ w2, const float* __restrict__ b2,
                               const float* __restrict__ w3, const float* __restrict__ b3,
                               float* __restrict__ logit) {
  __shared__ float h1s[8][16 * WSTRIDE];
  __shared__ float h2s[8][16 * H2S];
  __shared__ float W2t[32 * WSTRIDE];  // padded [col<32][k<64] of w2[50,25]
  int tid = threadIdx.x;
  for (int i = tid; i < 32 * WSTRIDE; i += 256) W2t[i] = 0.f;
  for (int j = tid; j < 50 * 25; j += 256) {
    int k = j / 25, c = j % 25;
    async_g2l_b32(&W2t[c * WSTRIDE + k], w2 + j);
  }
  int wib = tid >> 5;
  int wave = blockIdx.x * (blockDim.x >> 5) + wib;
  int lane = tid & 31;
  int half = lane >> 4, ml = lane & 15;
  int m0 = wave * 16;
  bool active = m0 < N_EDGES;
  float* h1 = h1s[wib];
  float* h2 = h2s[wib];

  if (active) {
    int rowa = m0 + ml;
    unsigned so = (unsigned)src[rowa] * 64u;
    unsigned dof = (unsigned)dst[rowa] * 64u;
    unsigned eo = (unsigned)rowa * 64u;
    v8f acc[4] = {};
#pragma unroll
    for (int kc = 0; kc < 3; ++kc) {
      v2f a[16];
#pragma unroll
      for (int ks = 0; ks < 16; ++ks) {
        int k = ks * 4 + 2 * half;
        float x0, x1;
        if (kc == 0)      { x0 = fmaxf(x[so + k], 0.f);  x1 = fmaxf(x[so + k + 1], 0.f); }
        else if (kc == 1) { x0 = fmaxf(x[dof + k], 0.f); x1 = fmaxf(x[dof + k + 1], 0.f); }
        else              { x0 = e[eo + k];              x1 = e[eo + k + 1]; }
        a[ks].x = x0; a[ks].y = x1;
      }
#pragma unroll
      for (int nt = 0; nt < 4; ++nt) {
        int col = nt * 16 + ml;
#pragma unroll
        for (int ks = 0; ks < 16; ++ks) {
          int k = kc * 64 + ks * 4 + 2 * half;
          v2f b;
          b.x = (col < 50) ? w1[(unsigned)k * 50 + col] : 0.f;
          b.y = (col < 50) ? w1[(unsigned)(k + 1) * 50 + col] : 0.f;
          acc[nt] = wmma_f32(a[ks], b, acc[nt]);
        }
      }
    }
#pragma unroll
    for (int nt = 0; nt < 4; ++nt) {
      int col = nt * 16 + ml;
      float bv = (col < 50) ? b1[col] : 0.f;
#pragma unroll
      for (int r = 0; r < 8; ++r)
        h1[(r + 8 * half) * WSTRIDE + col] = fmaxf(acc[nt][r] + bv, 0.f);
    }
  }
  wait_async0();
  __syncthreads();
  if (active) {
    v2f a[16];
#pragma unroll
    for (int ks = 0; ks < 16; ++ks)
      a[ks] = lds_ld2(&h1[ml * WSTRIDE + ks * 4 + 2 * half]);
#pragma unroll
    for (int nt = 0; nt < 2; ++nt) {
      int col = nt * 16 + ml;
      v8f acc = {};
#pragma unroll
      for (int ks = 0; ks < 16; ++ks) {
        v2f b = lds_ld2(&W2t[col * WSTRIDE + ks * 4 + 2 * half]);
        acc = wmma_f32(a[ks], b, acc);
      }
      float bv = (col < 25) ? b2[col] : 0.f;
#pragma unroll
      for (int r = 0; r < 8; ++r)
        h2[(r + 8 * half) * H2S + col] = fmaxf(acc[r] + bv, 0.f);
    }
  }
  __syncthreads();
  if (active && lane < 16) {
    float s = b3[0];
#pragma unroll
    for (int j = 0; j < 25; ++j) s += h2[ml * H2S + j] * w3[j];
    logit[m0 + ml] = s;
  }
}

// ---------------------------------------------------------------------------
extern "C" void kernel_launch(void* const* d_in, const int* in_sizes, int n_in,
                              void* d_out, int out_size, void* d_ws, size_t ws_size,
                              hipStream_t stream) {
  const float* x_in  = (const float*)d_in[0];
  const float* eattr = (const float*)d_in[1];
  const int*   src   = (const int*)d_in[2];
  const int*   dst   = (const int*)d_in[3];
  const float* nw    = (const float*)d_in[4];
  const float* nb    = (const float*)d_in[5];
  const float* ew    = (const float*)d_in[6];
  const float* eb    = (const float*)d_in[7];
  const float* linw  = (const float*)d_in[8];
  const float* linb  = (const float*)d_in[9];
  const float* cw1   = (const float*)d_in[10];
  const float* cb1   = (const float*)d_in[11];
  const float* cw2   = (const float*)d_in[12];
  const float* cb2   = (const float*)d_in[13];
  const float* ew1   = (const float*)d_in[14];
  const float* eb1   = (const float*)d_in[15];
  const float* ew2   = (const float*)d_in[16];
  const float* eb2   = (const float*)d_in[17];
  const float* gamma = (const float*)d_in[18];
  const float* beta  = (const float*)d_in[19];
  const float* mw1   = (const float*)d_in[20];
  const float* mb1   = (const float*)d_in[21];
  const float* mw2   = (const float*)d_in[22];
  const float* mb2   = (const float*)d_in[23];
  const float* mw3   = (const float*)d_in[24];
  const float* mb3   = (const float*)d_in[25];

  float* ws = (float*)d_ws;
  float* X1    = ws;
  float* X2    = X1 + (size_t)N_NODES * 64;
  float* E     = X2 + (size_t)N_NODES * 64;
  float* AGG   = E + (size_t)N_EDGES * 64;
  float* Hb    = AGG + (size_t)N_NODES * 64;
  float* STATS = Hb + (size_t)N_NODES * 64;

  float* xfinal = (float*)d_out;                  // [N,64]
  float* logit  = xfinal + (size_t)N_NODES * 64;  // [E]

  const int TPB = 256;
  const int gridN = (N_NODES / 16 + 7) / 8;  // 782
  const int gridE = (N_EDGES / 16 + 7) / 8;  // 5000

  gemm64_kernel<32><<<gridN, TPB, 0, stream>>>(x_in, nw, nb, X1, N_NODES);
  gemm64_kernel<16><<<gridE, TPB, 0, stream>>>(eattr, ew, eb, E, N_EDGES);

  const float* xcur = X1;
  for (int l = 0; l < NLAYER; ++l) {
    float* xnext = (l == NLAYER - 1) ? xfinal : X2;
    fill_zero_kernel<<<1024, TPB, 0, stream>>>((float4*)AGG, (size_t)N_NODES * 16);
    fill_zero_kernel<<<1, 32, 0, stream>>>((float4*)STATS, 32);
    msg_scatter_kernel<<<gridE, TPB, 0, stream>>>(xcur, E, src, dst,
        linw + (size_t)l * 64 * 64, linb + l * 64, AGG);
    conv_fused_kernel<<<gridN, TPB, 0, stream>>>(xcur, AGG,
        cw1 + (size_t)l * 64 * 64, cb1 + l * 64,
        cw2 + (size_t)l * 64 * 64, cb2 + l * 64, Hb, STATS);
    bn_apply_kernel<<<(int)(((size_t)N_NODES * 16 + TPB - 1) / TPB), TPB, 0, stream>>>(
        xcur, Hb, STATS, gamma + l * 64, beta + l * 64, xnext);
    edge_update_kernel<<<gridE, TPB, 0, stream>>>(xnext, E, src, dst,
        ew1 + (size_t)l * 192 * 64, eb1 + l * 64, ew2 + (size_t)l * 64 * 64, eb2 + l * 64);
    xcur = xnext;
  }

  readout_kernel<<<gridE, TPB, 0, stream>>>(xcur, E, src, dst,
      mw1, mb1, mw2, mb2, mw3, mb3, logit);
}